// Model_58420145160291
// MI455X (gfx1250) — compile-verified
//
#include <hip/hip_runtime.h>

// Problem constants (from reference)
#define SEQ_LEN 720
#define ENC_IN  862
#define PERIOD  24
#define NSEG    30     // 720/24
#define PAD     12
#define KW      25     // conv taps (residual folded in: h'[12] += 1)
// Tiling
#define CT      16     // series (c) per workgroup
#define TB      16     // time block (WMMA M)
#define NTB     45     // 720/16
#define KDIM    40     // GEMM K = TB + KW - 1
#define ROWP    748    // LDS row pitch (dwords): even (8B align) and 748%64=44, gcd(44,64)=4 -> conflict-free b64
#define MAXTILE 6      // ceil(45/8) time blocks per wave

typedef __attribute__((ext_vector_type(2))) float v2f;
typedef __attribute__((ext_vector_type(8))) float v8f;

// Build banded conv matrix A[16][40]: A[i][k] = h'[k-i], h' = conv_w with +1 at center.
__global__ __launch_bounds__(256) void build_A_kernel(const float* __restrict__ w,
                                                      float* __restrict__ A) {
  for (int idx = threadIdx.x; idx < TB * KDIM; idx += 256) {
    int i = idx / KDIM;
    int k = idx - i * KDIM;
    int j = k - i;
    float v = (j >= 0 && j < KW) ? w[j] : 0.0f;
    if (j == PAD) v += 1.0f;   // residual identity tap
    A[idx] = v;
  }
}

__global__ __launch_bounds__(256) void tsf_kernel(const float* __restrict__ x,
                                                  const float* __restrict__ Aw,
                                                  float* __restrict__ out) {
  __shared__ __align__(16) float z[CT][ROWP];     // zero halo [0,12), data [12,732), zero [732,748)
  __shared__ float meanv[CT];
  __shared__ float red[CT][17];
  __shared__ float S[NSEG][2][CT];                // per-segment parity sums of conv output

  const int tid = threadIdx.x;
  const int b   = blockIdx.y;
  const int c0  = blockIdx.x * CT;

  // ---- zero LDS (halos + sums) ----
  for (int i = tid; i < CT * ROWP; i += 256) ((float*)z)[i] = 0.0f;
  for (int i = tid; i < NSEG * 2 * CT; i += 256) ((float*)S)[i] = 0.0f;
  __syncthreads();

  // ---- stage tile: x is (B, L, C); consecutive lanes <-> consecutive c (coalesced 64B rows) ----
  for (int i = tid; i < CT * SEQ_LEN; i += 256) {
    int c = i & (CT - 1);
    int t = i >> 4;
    int gc = c0 + c;
    float v = 0.0f;
    if (gc < ENC_IN) v = x[((size_t)b * SEQ_LEN + t) * ENC_IN + gc];
    z[c][PAD + t] = v;
  }
  __syncthreads();

  // ---- per-series mean ----
  {
    int s = tid >> 4, u = tid & 15;
    float p = 0.0f;
    for (int t = u; t < SEQ_LEN; t += 16) p += z[s][PAD + t];
    red[s][u] = p;
  }
  __syncthreads();
  if (tid < CT) {
    float m = 0.0f;
    for (int u = 0; u < 16; ++u) m += red[tid][u];
    meanv[tid] = m * (1.0f / SEQ_LEN);
  }
  __syncthreads();

  // ---- subtract mean (halos stay zero -> correct 'same' zero padding of (x - mean)) ----
  for (int i = tid; i < CT * SEQ_LEN; i += 256) {
    int c = i & (CT - 1);
    int t = i >> 4;
    z[c][PAD + t] -= meanv[c];
  }
  __syncthreads();

  // ---- conv as GEMM on v_wmma_f32_16x16x4_f32: Y(16t x 16c) = A(16x40) * Zwin(40x16) ----
  const int wave = tid >> 5;
  const int lane = tid & 31;
  const int n = lane & 15;     // series column (B/C/D N index) == A M index
  const int h = lane >> 4;     // K half-select per ISA A/B layout

  v2f afr[10];                 // A fragments, hoisted (shared across all time blocks)
#pragma unroll
  for (int ks = 0; ks < 10; ++ks)
    afr[ks] = *(const v2f*)(Aw + n * KDIM + ks * 4 + 2 * h);

  v8f accs[MAXTILE];
#pragma unroll
  for (int it = 0; it < MAXTILE; ++it) {
    int tb = wave + it * 8;          // wave-uniform -> EXEC stays all-ones around WMMA
    if (tb >= NTB) break;
    int t0 = tb * TB;
    v8f acc = {0.f, 0.f, 0.f, 0.f, 0.f, 0.f, 0.f, 0.f};
#pragma unroll
    for (int ks = 0; ks < 10; ++ks) {
      // B[k][n] = z[n][t0 + k] (padded-row offset already includes the -12 window shift)
      v2f bfr = *(const v2f*)&z[n][t0 + ks * 4 + 2 * h];
      acc = __builtin_amdgcn_wmma_f32_16x16x4_f32(
          false, afr[ks], false, bfr, (short)0, acc, false, false);
    }
    accs[it] = acc;
    // D layout: acc[r] = v(t0 + 8h + r) of series n -> accumulate parity sums per segment
    int tbase = t0 + 8 * h;
#pragma unroll
    for (int r = 0; r < 8; ++r) {
      int t = tbase + r;
      atomicAdd(&S[t / PERIOD][t & 1][n], acc[r]);   // ds_add_f32
    }
  }
  __syncthreads();

  // ---- FFT low-pass (analytic): y = 0.5*v + (1/24)*sum(opposite parity in segment) + mean ----
  const int gc = c0 + n;
#pragma unroll
  for (int it = 0; it < MAXTILE; ++it) {
    int tb = wave + it * 8;
    if (tb >= NTB) break;
    int tbase = tb * TB + 8 * h;
#pragma unroll
    for (int r = 0; r < 8; ++r) {
      int t = tbase + r;
      float y = 0.5f * accs[it][r]
              + (1.0f / PERIOD) * S[t / PERIOD][(t & 1) ^ 1][n]
              + meanv[n];
      if (gc < ENC_IN)
        out[((size_t)b * SEQ_LEN + t) * ENC_IN + gc] = y;   // coalesced over c
    }
  }
}

extern "C" void kernel_launch(void* const* d_in, const int* in_sizes, int n_in,
                              void* d_out, int out_size, void* d_ws, size_t ws_size,
                              hipStream_t stream) {
  const float* x = (const float*)d_in[0];      // (B, 720, 862) f32
  const float* w = (const float*)d_in[1];      // (1,1,25) f32
  float* outp = (float*)d_out;
  float* Aws  = (float*)d_ws;                  // 16*40 floats = 2560 B

  const int B = in_sizes[0] / (SEQ_LEN * ENC_IN);

  build_A_kernel<<<1, 256, 0, stream>>>(w, Aws);
  dim3 grid((ENC_IN + CT - 1) / CT, B);
  tsf_kernel<<<grid, 256, 0, stream>>>(x, Aws, outp);
}